// SwitchReverseTriu_12962211299498
// MI455X (gfx1250) — compile-verified
//
#include <hip/hip_runtime.h>

// ---------------------------------------------------------------------------
// SwitchReverseTriu for MI455X (gfx1250).
//
// out[b,h,i] = rev[b] ? x[b,h,N-1-i] : x[b,h,i],  N = 99681, fp32, 16x32 rows.
//
// Roofline: 0 FLOPs, ~0.408 GB moved (0.204 GB read + 0.204 GB write).
// At 23.3 TB/s HBM the floor is ~17.5 us; purely bandwidth bound, so the goal
// is full-cacheline transactions + high MLP. WMMA is irrelevant (no matmul).
// The CDNA5-specific path for this op is the async global<->LDS DMA engine
// (GLOBAL_LOAD_ASYNC_TO_LDS_* / GLOBAL_STORE_ASYNC_FROM_LDS_*, ASYNCcnt):
// data streams memory -> LDS -> memory without a VGPR round trip.
//
// Alignment: N % 4 == 1, so row r starts at element offset r (mod 4).
// We align the *store* side (i0 = (-row) mod 4) and emit 16B b128 stores.
// A reversed source group can never also be 16B aligned, so the reversed
// branch uses four 4B async loads (consecutive descending addresses -> the
// cache still sees full 128B lines). Row head/tail (<=3 elems) is scalar.
// ---------------------------------------------------------------------------

#define UT_LEN 99681   // (448-2)*(448-2+1)/2
#define NROWS  512     // 16 batches * 32 heads
#define BLOCK  256     // 8 waves (wave32)

typedef float v4f __attribute__((ext_vector_type(4)));
// Must match the builtin's parameter type exactly: 4 x i32, gcc vector_size.
typedef int v4i __attribute__((vector_size(4 * sizeof(int))));

#if defined(__AMDGCN__) &&                                                    \
    __has_builtin(__builtin_amdgcn_global_load_async_to_lds_b32) &&           \
    __has_builtin(__builtin_amdgcn_global_load_async_to_lds_b128) &&          \
    __has_builtin(__builtin_amdgcn_global_store_async_from_lds_b128) &&       \
    __has_builtin(__builtin_amdgcn_s_wait_asynccnt)
#define HAS_ASYNC 1
#else
#define HAS_ASYNC 0
#endif

#if HAS_ASYNC
typedef __attribute__((address_space(1))) int*  g32_t;   // global i32
typedef __attribute__((address_space(1))) v4i*  g128_t;  // global 4xi32
typedef __attribute__((address_space(3))) int*  l32_t;   // LDS i32
typedef __attribute__((address_space(3))) v4i*  l128_t;  // LDS 4xi32
#endif

__global__ __launch_bounds__(BLOCK) void
switch_reverse_triu_kernel(const float* __restrict__ x,
                           const unsigned char* __restrict__ rev,  // jnp.bool_: 1 byte/elem
                           float* __restrict__ out) {
  const int N   = UT_LEN;
  const int row = blockIdx.y;                 // row = b*32 + h
  const bool r  = rev[row >> 5] != 0;
  const long long base = (long long)row * N;

  // (row*N) % 4 == row % 4, so first 16B-aligned output element is i0.
  const int i0   = (4 - (row & 3)) & 3;
  const int nvec = (N - i0) >> 2;             // number of float4 groups
  const int v    = blockIdx.x * BLOCK + threadIdx.x;

#if HAS_ASYNC
  __shared__ __align__(16) char smem[BLOCK * 16];  // one 16B slot per lane
#endif

  if (v < nvec) {
    const int    i   = i0 + (v << 2);
    const float* src = x + base;
    float*       dst = out + base + i;        // 16B aligned by construction

#if HAS_ASYNC
    // Lane-private LDS slot: the b128 store reads only bytes this lane's own
    // async loads wrote, so s_wait_asynccnt 0 is the only sync needed.
    char* slot = smem + threadIdx.x * 16;
    if (!r) {
      __builtin_amdgcn_global_load_async_to_lds_b128((g128_t)(src + i),
                                                     (l128_t)slot, 0, 0);
    } else {
      const int j = N - 1 - i;                // out[i+m] = src[j-m]
      __builtin_amdgcn_global_load_async_to_lds_b32((g32_t)(src + j    ),
                                                    (l32_t)(slot     ), 0, 0);
      __builtin_amdgcn_global_load_async_to_lds_b32((g32_t)(src + j - 1),
                                                    (l32_t)(slot +  4), 0, 0);
      __builtin_amdgcn_global_load_async_to_lds_b32((g32_t)(src + j - 2),
                                                    (l32_t)(slot +  8), 0, 0);
      __builtin_amdgcn_global_load_async_to_lds_b32((g32_t)(src + j - 3),
                                                    (l32_t)(slot + 12), 0, 0);
    }
    __builtin_amdgcn_s_wait_asynccnt(0);
    __builtin_amdgcn_global_store_async_from_lds_b128((g128_t)dst,
                                                      (l128_t)slot, 0, 0);
#else
    // Fallback: direct VGPR path with non-temporal hints (streaming data,
    // don't pollute L2 twice).
    if (!r) {
      v4f d = __builtin_nontemporal_load((const v4f*)(src + i));
      __builtin_nontemporal_store(d, (v4f*)dst);
    } else {
      const int j = N - 1 - i;
      v4f d;
      d.x = __builtin_nontemporal_load(src + j);
      d.y = __builtin_nontemporal_load(src + j - 1);
      d.z = __builtin_nontemporal_load(src + j - 2);
      d.w = __builtin_nontemporal_load(src + j - 3);
      __builtin_nontemporal_store(d, (v4f*)dst);
    }
#endif
  }

  // Per-row head (i < i0) and tail (< 4) cleanup: <=6 scalar elements.
  if (blockIdx.x == 0 && threadIdx.x == 0) {
    for (int i = 0; i < i0; ++i)
      out[base + i] = r ? x[base + N - 1 - i] : x[base + i];
    for (int i = i0 + (nvec << 2); i < N; ++i)
      out[base + i] = r ? x[base + N - 1 - i] : x[base + i];
  }
}

extern "C" void kernel_launch(void* const* d_in, const int* in_sizes, int n_in,
                              void* d_out, int out_size, void* d_ws, size_t ws_size,
                              hipStream_t stream) {
  const float*         x   = (const float*)d_in[0];
  const unsigned char* rev = (const unsigned char*)d_in[1];  // jnp.bool_ -> 1 byte
  float*               out = (float*)d_out;

  const int nvecMax = UT_LEN >> 2;                         // 24920 groups max/row
  dim3 grid((nvecMax + BLOCK - 1) / BLOCK, NROWS);         // (98, 512)
  switch_reverse_triu_kernel<<<grid, dim3(BLOCK), 0, stream>>>(x, rev, out);
}